// AttentionLayer_13262859010765
// MI455X (gfx1250) — compile-verified
//
#include <hip/hip_runtime.h>
#include <cmath>

#define B_  4
#define TE_ 1024
#define TD_ 512
#define H_  128

typedef float v2f __attribute__((ext_vector_type(2)));
typedef float v8f __attribute__((ext_vector_type(8)));

// ---------------------------------------------------------------------------
// Fast tanh: native gfx1250 v_tanh_f32 if available, else exp-based (safe at
// large |x| via copysign form).
// ---------------------------------------------------------------------------
__device__ __forceinline__ float fast_tanh(float x) {
#if __has_builtin(__builtin_amdgcn_tanhf)
  return __builtin_amdgcn_tanhf(x);
#else
  float ax = fabsf(x);
  float e2 = __expf(2.0f * ax);
  float t  = 1.0f - 2.0f / (e2 + 1.0f);
  return copysignf(t, x);
#endif
}

// ---------------------------------------------------------------------------
// Generic batched f32 GEMM via V_WMMA_F32_16X16X4_F32 (wave32, 1 wave = one
// 16x16 C tile). Row-major A[M x K] (lda=K), B[K x N] (ldb=N), C[M x N].
// Dual accumulators break the WMMA D->C dependency chain (K % 8 == 0 here).
//   grid.x = N/16, grid.y = M/16, grid.z = batch; block = 32 threads.
// ---------------------------------------------------------------------------
__global__ void __launch_bounds__(32)
wmma_gemm_f32_kernel(const float* __restrict__ A, const float* __restrict__ Bm,
                     float* __restrict__ C, int N, int K,
                     long long sA, long long sB, long long sC) {
  const int tn = blockIdx.x, tm = blockIdx.y, bz = blockIdx.z;
  const float* Ab = A  + (long long)bz * sA;
  const float* Bb = Bm + (long long)bz * sB;
  float*       Cb = C  + (long long)bz * sC;

  const int lane = threadIdx.x;      // 0..31, EXEC all ones (no divergence)
  const int half = lane >> 4;        // 0 or 1
  const int l16  = lane & 15;
  const long long arow = (long long)(tm * 16 + l16) * K;  // A row this lane feeds
  const int bcol = tn * 16 + l16;                         // B col this lane feeds

  v8f acc0 = {};
  v8f acc1 = {};
  for (int k0 = 0; k0 < K; k0 += 8) {
    // chunk k0 .. k0+3
    {
      const int ka = k0 + half * 2;
      v2f a, b;
      a.x = Ab[arow + ka];
      a.y = Ab[arow + ka + 1];
      b.x = Bb[(long long)ka * N + bcol];
      b.y = Bb[(long long)(ka + 1) * N + bcol];
      acc0 = __builtin_amdgcn_wmma_f32_16x16x4_f32(
          /*neg_a=*/false, a, /*neg_b=*/false, b,
          /*c_mod=*/(short)0, acc0, /*reuse_a=*/false, /*reuse_b=*/false);
    }
    // chunk k0+4 .. k0+7
    {
      const int ka = k0 + 4 + half * 2;
      v2f a, b;
      a.x = Ab[arow + ka];
      a.y = Ab[arow + ka + 1];
      b.x = Bb[(long long)ka * N + bcol];
      b.y = Bb[(long long)(ka + 1) * N + bcol];
      acc1 = __builtin_amdgcn_wmma_f32_16x16x4_f32(
          false, a, false, b, (short)0, acc1, false, false);
    }
  }
  // D layout: element j of acc -> C[tm*16 + j + 8*half][tn*16 + l16]
  #pragma unroll
  for (int j = 0; j < 8; ++j) {
    const int m = tm * 16 + j + 8 * half;
    Cb[(long long)m * N + bcol] = acc0[j] + acc1[j];
  }
}

// ---------------------------------------------------------------------------
// Fused energies + softmax. One block (256 threads = 8 waves) per (b, d).
//   e_raw[t] = sum_h Va[h] * tanh(Ws[b,t,h] + Uh[b,d,h]),  t = 0..TE-1
//   e_out[b,d,:] = softmax_t(e_raw)
// Per-lane float4 over h (lane l owns h = 4l..4l+3), wave32 shuffle reduce.
// ---------------------------------------------------------------------------
__global__ void __launch_bounds__(256)
energies_softmax_kernel(const float* __restrict__ Ws,
                        const float* __restrict__ Uh,
                        const float* __restrict__ Va,
                        float* __restrict__ e_out) {
  __shared__ float s_e[TE_];
  __shared__ float s_red[8];

  const int bd   = blockIdx.x;          // 0 .. B*TD-1
  const int b    = bd / TD_;
  const int tid  = threadIdx.x;         // 0..255
  const int lane = tid & 31;
  const int wave = tid >> 5;            // 0..7

  const float4 uh = ((const float4*)(Uh + (long long)bd * H_))[lane];
  const float4 v  = ((const float4*)Va)[lane];
  const float4* __restrict__ wsb =
      (const float4*)(Ws + (long long)b * TE_ * H_);

  const int t0 = wave * (TE_ / 8);
  const int t1 = t0 + (TE_ / 8);
  for (int t = t0; t < t1; ++t) {
    __builtin_prefetch(wsb + (long long)(t + 4) * (H_ / 4) + lane, 0, 1);
    const float4 w = wsb[(long long)t * (H_ / 4) + lane];
    float s = v.x * fast_tanh(w.x + uh.x)
            + v.y * fast_tanh(w.y + uh.y)
            + v.z * fast_tanh(w.z + uh.z)
            + v.w * fast_tanh(w.w + uh.w);
    #pragma unroll
    for (int off = 16; off > 0; off >>= 1) s += __shfl_xor(s, off, 32);
    if (lane == 0) s_e[t] = s;
  }
  __syncthreads();

  // --- block max ---
  float m = -3.402823466e38f;
  for (int i = tid; i < TE_; i += 256) m = fmaxf(m, s_e[i]);
  #pragma unroll
  for (int off = 16; off > 0; off >>= 1) m = fmaxf(m, __shfl_xor(m, off, 32));
  if (lane == 0) s_red[wave] = m;
  __syncthreads();
  m = s_red[0];
  #pragma unroll
  for (int w2 = 1; w2 < 8; ++w2) m = fmaxf(m, s_red[w2]);

  // --- block sum of exp ---
  float sum = 0.0f;
  for (int i = tid; i < TE_; i += 256) sum += __expf(s_e[i] - m);
  #pragma unroll
  for (int off = 16; off > 0; off >>= 1) sum += __shfl_xor(sum, off, 32);
  __syncthreads();                       // done reading s_red before overwrite
  if (lane == 0) s_red[wave] = sum;
  __syncthreads();
  sum = 0.0f;
  #pragma unroll
  for (int w2 = 0; w2 < 8; ++w2) sum += s_red[w2];
  const float inv = 1.0f / sum;

  float* __restrict__ eo = e_out + (long long)bd * TE_;
  for (int i = tid; i < TE_; i += 256) eo[i] = __expf(s_e[i] - m) * inv;
}

// ---------------------------------------------------------------------------
// Launch: inputs (dict order) = encoder_out, decoder_out, W_a, U_a, V_a.
// d_out = [ c (B*TD*H) | e (B*TD*TE) ]  (tuple return order).
// d_ws  = [ Ws (B*TE*H f32) | Uh (B*TD*H f32) ]  = 3 MB.
// ---------------------------------------------------------------------------
extern "C" void kernel_launch(void* const* d_in, const int* in_sizes, int n_in,
                              void* d_out, int out_size, void* d_ws, size_t ws_size,
                              hipStream_t stream) {
  const float* enc = (const float*)d_in[0];
  const float* dec = (const float*)d_in[1];
  const float* Wa  = (const float*)d_in[2];
  const float* Ua  = (const float*)d_in[3];
  const float* Va  = (const float*)d_in[4];

  float* c_out = (float*)d_out;
  float* e_out = c_out + (size_t)B_ * TD_ * H_;

  float* Ws = (float*)d_ws;                       // [B*TE, H]
  float* Uh = Ws + (size_t)B_ * TE_ * H_;         // [B*TD, H]

  // Phase 1a: Ws = enc @ W_a   (M = B*TE fused: shared weights across batch)
  {
    dim3 grid(H_ / 16, (B_ * TE_) / 16, 1);
    wmma_gemm_f32_kernel<<<grid, 32, 0, stream>>>(enc, Wa, Ws, H_, H_, 0, 0, 0);
  }
  // Phase 1b: Uh = dec @ U_a
  {
    dim3 grid(H_ / 16, (B_ * TD_) / 16, 1);
    wmma_gemm_f32_kernel<<<grid, 32, 0, stream>>>(dec, Ua, Uh, H_, H_, 0, 0, 0);
  }
  // Phase 2: energies + softmax -> e_out
  {
    dim3 grid(B_ * TD_, 1, 1);
    energies_softmax_kernel<<<grid, 256, 0, stream>>>(Ws, Uh, Va, e_out);
  }
  // Phase 3: c[b] = e[b] @ enc[b]   (batched, K = TE)
  {
    dim3 grid(H_ / 16, TD_ / 16, B_);
    wmma_gemm_f32_kernel<<<grid, 32, 0, stream>>>(
        e_out, enc, c_out, H_, TE_,
        (long long)TD_ * TE_, (long long)TE_ * H_, (long long)TD_ * H_);
  }
}